// HomeostaticFieldHardened_25615184953649
// MI455X (gfx1250) — compile-verified
//
#include <hip/hip_runtime.h>
#include <math.h>

// ---------------------------------------------------------------------------
// Hyperbolic homeostatic field update, MI455X (gfx1250, wave32).
//
// Memory-bound streaming kernel (~520MB @ 23.3TB/s => ~22us floor).
// The 1Mx65 @ 65x16 Minkowski Gram matrix runs on the matrix pipe with
// V_WMMA_F32_16X16X4_F32 (fp32 precision preserved; anchors are clustered at
// 0.01 scale so low precision could flip argmin). All global traffic is a
// single coalesced read + coalesced write pass staged through LDS, with the
// load loop unrolled x13 for deep memory-level parallelism (loads are made
// unconditional via address clamping so they pipeline).
// ---------------------------------------------------------------------------

#define ALPHA 0.1f
#define EPSF  1e-7f

typedef float v2f __attribute__((ext_vector_type(2)));
typedef float v8f __attribute__((ext_vector_type(8)));

constexpr int DD        = 65;   // 1 time + 64 space
constexpr int NA        = 16;   // anchors
constexpr int PPB       = 128;  // points per block (4 waves, 32 pts/wave)
constexpr int KPAD      = 68;   // 17 * 4 (K padded for WMMA x4 steps)
constexpr int XS_STRIDE = 69;   // odd stride -> conflict-free columns
constexpr int IN_STRIDE = 17;   // inner[point][anchor] stride

__global__ __launch_bounds__(PPB)
void hyper_field_kernel(const float* __restrict__ g_x,
                        const float* __restrict__ g_anch,
                        float* __restrict__ g_out,
                        int B)
{
    __shared__ float xs[PPB * XS_STRIDE];       // point tile, row-major
    __shared__ float anch[NA * KPAD];           // anchors, time NEGATED, K-padded
    __shared__ float innerL[PPB * IN_STRIDE];   // Gram tile
    __shared__ float pScale[PPB], pXY[PPB], pCh[PPB], pSinhc[PPB];
    __shared__ int   pIdx[PPB];

    const int  tid           = threadIdx.x;
    const int  blockBase     = blockIdx.x * PPB;
    const long totalElems    = (long)B * DD;
    const long blockElemBase = (long)blockBase * DD;

    // ---- Phase 0: project anchors into LDS (time recomputed & negated) ----
    // Fully unrolled so all 64 anchor loads are in flight at once (wave 0 does
    // this right before the block-wide barrier).
    if (tid < NA) {
        float ssum = 0.f;
        #pragma unroll
        for (int k = 1; k < DD; ++k) {
            float a = g_anch[tid * DD + k];
            anch[tid * KPAD + k] = a;
            ssum += a * a;
        }
        anch[tid * KPAD + 0]  = -sqrtf(1.f + ssum);  // negated time component
        anch[tid * KPAD + 65] = 0.f;
        anch[tid * KPAD + 66] = 0.f;
        anch[tid * KPAD + 67] = 0.f;
    }

    // ---- Phase 1: fully-coalesced streaming load of 128x65 tile into LDS ----
    // j = i*128 + tid over the contiguous tile; track p = j/65, k = j%65
    // incrementally (128 = 1*65 + 63). Loads are UNCONDITIONAL (address
    // clamped to the last valid element) so they pipeline; tail rows hold
    // duplicate data but their outputs are never stored.
    {
        int p = (tid >= DD) ? 1 : 0;
        int k = tid - p * DD;
        #pragma unroll 13
        for (int i = 0; i < DD; ++i) {
            long e = blockElemBase + (long)i * PPB + tid;
            if (e >= totalElems) e = totalElems - 1;   // clamp, not branch-skip
            xs[p * XS_STRIDE + k] = g_x[e];
            p += 1; k += 63;
            if (k >= DD) { k -= DD; p += 1; }
        }
        // zero-pad K columns 65..67 of own row
        xs[tid * XS_STRIDE + 65] = 0.f;
        xs[tid * XS_STRIDE + 66] = 0.f;
        xs[tid * XS_STRIDE + 67] = 0.f;
    }
    __syncthreads();

    // ---- Phase 2: Gram matrix via V_WMMA_F32_16X16X4_F32 ----
    // Wave w handles points [32w, 32w+32): two 16x16 tiles, interleaved
    // accumulation chains. A-fragment layout (f32 16x4): lanes 0-15 hold
    // K={kk,kk+1}, lanes 16-31 hold K={kk+2,kk+3}; B mirrored over N=lane%16.
    const int wave = tid >> 5;
    const int lane = tid & 31;
    const int ln   = lane & 15;
    const int h    = lane >> 4;          // half-wave select (K pair)

    const float* xrow0 = &xs[(wave * 32 + ln)      * XS_STRIDE];
    const float* xrow1 = &xs[(wave * 32 + 16 + ln) * XS_STRIDE];
    const float* brow  = &anch[ln * KPAD];

    v8f c0 = {};
    v8f c1 = {};
    for (int kk = 0; kk < KPAD; kk += 4) {
        const int kb = kk + 2 * h;
        v2f a0 = { xrow0[kb], xrow0[kb + 1] };
        v2f a1 = { xrow1[kb], xrow1[kb + 1] };
        v2f b  = { brow[kb],  brow[kb + 1]  };
        c0 = __builtin_amdgcn_wmma_f32_16x16x4_f32(
                 false, a0, false, b, (short)0, c0, false, false);
        c1 = __builtin_amdgcn_wmma_f32_16x16x4_f32(
                 false, a1, false, b, (short)0, c1, false, false);
    }

    // Spill C tiles to LDS as inner[point][anchor].
    // C layout: VGPR r -> row M=r (lanes 0-15) / M=8+r (lanes 16-31), N=lane%16.
    #pragma unroll
    for (int r = 0; r < 8; ++r) {
        const int pt0 = wave * 32 + r + 8 * h;
        innerL[pt0 * IN_STRIDE + ln]        = c0[r];
        innerL[(pt0 + 16) * IN_STRIDE + ln] = c1[r];
    }
    // innerL rows for this wave are written and read by the same wave only:
    // DS ops are in-order within a wave, no workgroup barrier needed here.

    // ---- Phase 3: per-point argmin + scalar exponential-map params ----
    {
        float bestZ = INFINITY;
        int   bestN = 0;
        float xy    = 0.f;
        #pragma unroll
        for (int n = 0; n < NA; ++n) {
            float in = innerL[tid * IN_STRIDE + n];
            float z  = fmaxf(-in, 1.f + EPSF);       // clamped like reference
            if (z < bestZ) { bestZ = z; bestN = n; xy = in; }  // first-tie wins
        }

        float d = acoshf(fmaxf(-xy, 1.f + EPSF));

        // linner(u,u), u = na + xy*x  (na time = -anch[..][0], anch stored negated)
        const float* xr = &xs[tid * XS_STRIDE];
        const float* ar = &anch[bestN * KPAD];
        float u0  = -ar[0] + xy * xr[0];
        float luu = -(u0 * u0);
        #pragma unroll 4
        for (int k = 1; k < DD; ++k) {
            float u = ar[k] + xy * xr[k];
            luu += u * u;
        }

        float unorm = sqrtf(fmaxf(luu, EPSF));
        float scale = ALPHA * d / unorm;
        float lvv   = scale * scale * luu;
        float vn    = sqrtf(fmaxf(lvv, EPSF * EPSF));
        float ex    = expf(vn);
        float exi   = 1.f / ex;
        float ch    = 0.5f * (ex + exi);
        float sh    = 0.5f * (ex - exi);
        float sinhc = (vn > EPSF) ? (sh / vn) : 1.f;

        pScale[tid] = scale;
        pXY[tid]    = xy;
        pCh[tid]    = ch;
        pSinhc[tid] = sinhc;
        pIdx[tid]   = bestN;
    }
    __syncthreads();

    // ---- Phase 4: fully-coalesced streaming store ----
    {
        int p = (tid >= DD) ? 1 : 0;
        int k = tid - p * DD;
        #pragma unroll 5
        for (int i = 0; i < DD; ++i) {
            long e = blockElemBase + (long)i * PPB + tid;
            if (e < totalElems) {
                float x  = xs[p * XS_STRIDE + k];
                int   nn = pIdx[p];
                float av = anch[nn * KPAD + k];
                float na = (k == 0) ? -av : av;      // un-negate time component
                float v  = pScale[p] * (na + pXY[p] * x);
                g_out[e] = pCh[p] * x + pSinhc[p] * v;
            }
            p += 1; k += 63;
            if (k >= DD) { k -= DD; p += 1; }
        }
    }
}

// ---------------------------------------------------------------------------
extern "C" void kernel_launch(void* const* d_in, const int* in_sizes, int n_in,
                              void* d_out, int out_size, void* d_ws, size_t ws_size,
                              hipStream_t stream)
{
    const float* g_x    = (const float*)d_in[0];   // hyp_emb, (B, 65) f32
    const float* g_anch = (const float*)d_in[1];   // anchors, (16, 65) f32
    float*       g_out  = (float*)d_out;           // (B, 65) f32

    const int B    = in_sizes[0] / DD;
    const int grid = (B + PPB - 1) / PPB;

    hipLaunchKernelGGL(hyper_field_kernel, dim3(grid), dim3(PPB), 0, stream,
                       g_x, g_anch, g_out, B);
}